// YoloLayer_67388036874753
// MI455X (gfx1250) — compile-verified
//
#include <hip/hip_runtime.h>

// ---------------------------------------------------------------------------
// YOLO head, MI455X (gfx1250). Pure bandwidth problem: ~213 MB traffic,
// ~9.1 us roofline at 23.3 TB/s. Strategy: 128-bit coalesced loads,
// LDS-staged transpose so the AoS (.,12) output is written with fully
// coalesced b128 stores via gfx1250 async LDS->global stores (ASYNCcnt).
// ---------------------------------------------------------------------------

#define YG      152
#define YGG     (YG * YG)          // 23104
#define YNA     3
#define YATTR   12
#define YBLOCK  256
#define YEPT    4                  // elements per thread (consecutive grid cells)
#define YEPB    (YBLOCK * YEPT)    // 1024 elements per block
#define PAD4    4                  // pad dwords per 48-dword group (keeps 16B align)
#define LDS_DW  (YEPB * YATTR + (YEPB * YATTR / 48) * PAD4)   // 13312 dwords = 52 KB

#if defined(__HIP_DEVICE_COMPILE__)
#if __has_builtin(__builtin_amdgcn_global_store_async_from_lds_b128)
#define HAVE_ASYNC_ST 1
#endif
#if __has_builtin(__builtin_amdgcn_s_wait_asynccnt)
#define HAVE_ASYNC_WAIT 1
#endif
#endif

// int4 vector type + address-space-qualified variants for the async builtin
typedef int v4i __attribute__((ext_vector_type(4)));
typedef __attribute__((address_space(1))) v4i v4i_global;
typedef __attribute__((address_space(3))) v4i v4i_lds;

__device__ __forceinline__ float sigmoidf_fast(float v) {
    // matches jax.nn.sigmoid to ~1-2 ulp; v_exp_f32 fast path
    return 1.0f / (1.0f + __expf(-v));
}

__global__ __launch_bounds__(YBLOCK) void yolo_head_kernel(
    const float* __restrict__ x,
    const float* __restrict__ anchors,
    const int*   __restrict__ img_size_p,
    float*       __restrict__ out)
{
    __shared__ float lds[LDS_DW];

    const unsigned lt = threadIdx.x;
    const unsigned e0 = blockIdx.x * YEPB + lt * YEPT;     // flat element id (b,a,s)
    const unsigned plane = e0 / YGG;                       // b*3 + a
    const unsigned s0    = e0 - plane * YGG;               // multiple of 4, same row
    const unsigned a     = plane % YNA;
    const unsigned gy    = s0 / YG;
    const unsigned gx0   = s0 - gy * YG;

    const float stride = (float)(*img_size_p) / (float)YG; // 4.0 for 608/152
    const float ah = anchors[a * 5 + 0] / stride;
    const float aw = anchors[a * 5 + 1] / stride;
    const float al = anchors[a * 5 + 2] / stride;

    // 12 coalesced b128 loads (planes are YGG floats apart, all 16B aligned)
    const float* base = x + (size_t)plane * (YATTR * YGG) + s0;
    float4 q[YATTR];
#pragma unroll
    for (int k = 0; k < YATTR; ++k)
        q[k] = *(const float4*)(base + k * YGG);
    const float* qf = (const float*)q;   // qf[k*4 + i]

    const float gyf = (float)gy;
#pragma unroll
    for (int i = 0; i < YEPT; ++i) {
        const float gxf = (float)(gx0 + i);
        float o[YATTR];
        // keep reference rounding order: (sigmoid + grid) rounded, then * stride
        o[0]  = __fmul_rn(__fadd_rn(sigmoidf_fast(qf[0*4+i]), gxf), stride);
        o[1]  = __fmul_rn(__fadd_rn(sigmoidf_fast(qf[1*4+i]), gyf), stride);
        o[2]  = sigmoidf_fast(qf[2*4+i]);
        o[3]  = __fmul_rn(__fmul_rn(fminf(__expf(qf[3*4+i]), 1000.0f), ah), stride);
        o[4]  = __fmul_rn(__fmul_rn(fminf(__expf(qf[4*4+i]), 1000.0f), aw), stride);
        o[5]  = __fmul_rn(__fmul_rn(fminf(__expf(qf[5*4+i]), 1000.0f), al), stride);
        o[6]  = qf[6*4+i];
        o[7]  = qf[7*4+i];
        o[8]  = sigmoidf_fast(qf[8*4+i]);
        o[9]  = sigmoidf_fast(qf[9*4+i]);
        o[10] = sigmoidf_fast(qf[10*4+i]);
        o[11] = sigmoidf_fast(qf[11*4+i]);

        // padded LDS layout: flat f = lt*48 + i*12 + attr  ->  lt*52 + i*12 + attr
        float* dst = &lds[lt * 52 + i * 12];
#pragma unroll
        for (int c = 0; c < 3; ++c)
            *(float4*)(dst + 4 * c) =
                make_float4(o[4*c + 0], o[4*c + 1], o[4*c + 2], o[4*c + 3]);
    }

    __syncthreads();

    // Drain LDS tile to global with perfectly coalesced b128 stores.
    const size_t out_base = (size_t)blockIdx.x * (YEPB * YATTR);
#pragma unroll
    for (int j = 0; j < YATTR; ++j) {
        const unsigned f    = (unsigned)j * (YBLOCK * 4) + lt * 4; // 16B-lane-stride
        const unsigned lidx = f + (f / 48u) * PAD4;                // de-pad, 16B aligned
#if defined(HAVE_ASYNC_ST)
        __builtin_amdgcn_global_store_async_from_lds_b128(
            (v4i_global*)(out + out_base + f),
            (v4i_lds*)&lds[lidx],
            0, 0);
#else
        *(float4*)(out + out_base + f) = *(const float4*)&lds[lidx];
#endif
    }
#if defined(HAVE_ASYNC_ST) && defined(HAVE_ASYNC_WAIT)
    __builtin_amdgcn_s_wait_asynccnt(0);
#endif
}

extern "C" void kernel_launch(void* const* d_in, const int* in_sizes, int n_in,
                              void* d_out, int out_size, void* d_ws, size_t ws_size,
                              hipStream_t stream)
{
    const float* x       = (const float*)d_in[0];
    const float* anchors = (const float*)d_in[1];
    const int*   imgsz   = (const int*)d_in[2];
    float*       out     = (float*)d_out;

    // total (b,a,gy,gx) elements = in_sizes[0] / 12 = 2,217,984 = 2166 * 1024 exactly
    const int total_elems = in_sizes[0] / YATTR;
    const int blocks      = total_elems / YEPB;

    yolo_head_kernel<<<blocks, YBLOCK, 0, stream>>>(x, anchors, imgsz, out);
}